// ContextGNN_59030030516361
// MI455X (gfx1250) — compile-verified
//
#include <hip/hip_runtime.h>

#define IN_DIM 128
#define CD 64

typedef float v2f __attribute__((ext_vector_type(2)));
typedef float v8f __attribute__((ext_vector_type(8)));

__device__ __forceinline__ unsigned f2key(float f) {
    unsigned b = __float_as_uint(f);
    return (b & 0x80000000u) ? ~b : (b | 0x80000000u);
}
__device__ __forceinline__ float key2f(unsigned k) {
    unsigned b = (k & 0x80000000u) ? (k & 0x7fffffffu) : ~k;
    return __uint_as_float(b);
}
__device__ __forceinline__ float lrelu(float x) { return x > 0.f ? x : 0.01f * x; }

// ---- v_s = Wc^T @ w_s, v_d = Wc^T @ w_d  (fold Cu projection into 128-vectors) ----
__global__ void k_vsvd(const float* __restrict__ Wc, const float* __restrict__ aw,
                       float* __restrict__ vs, float* __restrict__ vd) {
    int t = threadIdx.x;             // 0..127
    float s = 0.f, d = 0.f;
    for (int c = 0; c < CD; ++c) {
        float w = Wc[c * IN_DIM + t];
        s += w * aw[c];
        d += w * aw[CD + c];
    }
    vs[t] = s;
    vd[t] = d;
}

// ---- fused: column-sum (for mean) + per-node attention scores s_s, s_d ----
__global__ __launch_bounds__(256) void k_stats(const float* __restrict__ h, int n,
                                               float* __restrict__ s_s, float* __restrict__ s_d,
                                               const float* __restrict__ vs,
                                               const float* __restrict__ vd,
                                               float* __restrict__ gsum) {
    const int lane = threadIdx.x & 31;
    const int wid  = (blockIdx.x * blockDim.x + threadIdx.x) >> 5;
    const int nw   = (gridDim.x * blockDim.x) >> 5;
    const float4 vs4 = *(const float4*)(vs + 4 * lane);
    const float4 vd4 = *(const float4*)(vd + 4 * lane);
    float a0 = 0.f, a1 = 0.f, a2 = 0.f, a3 = 0.f;
    for (int row = wid; row < n; row += nw) {
        const float4 x = *(const float4*)(h + (size_t)row * IN_DIM + 4 * lane);
        a0 += x.x; a1 += x.y; a2 += x.z; a3 += x.w;
        float ps = x.x * vs4.x + x.y * vs4.y + x.z * vs4.z + x.w * vs4.w;
        float pd = x.x * vd4.x + x.y * vd4.y + x.z * vd4.z + x.w * vd4.w;
        for (int off = 16; off; off >>= 1) {          // wave32 butterfly
            ps += __shfl_xor(ps, off);
            pd += __shfl_xor(pd, off);
        }
        if (lane == 0) { s_s[row] = ps; s_d[row] = pd; }
    }
    atomicAdd(&gsum[4 * lane + 0], a0);
    atomicAdd(&gsum[4 * lane + 1], a1);
    atomicAdd(&gsum[4 * lane + 2], a2);
    atomicAdd(&gsum[4 * lane + 3], a3);
}

// ---- tiny: global attention gates gA[4] from means ----
__global__ void k_ga(const float* __restrict__ gsumU, const float* __restrict__ gsumI,
                     const float* __restrict__ Wq, const float* __restrict__ Wk,
                     float* __restrict__ gA, float invNU, float invNI) {
    __shared__ float sqU[CD], skU[CD], sqI[CD], skI[CD];
    int c = threadIdx.x;             // 0..63
    float qu = 0.f, ku = 0.f, qi = 0.f, ki = 0.f;
    for (int t = 0; t < IN_DIM; ++t) {
        float u  = gsumU[t] * invNU;
        float it = gsumI[t] * invNI;
        float wq = Wq[c * IN_DIM + t];
        float wk = Wk[c * IN_DIM + t];
        qu += u * wq;  ku += u * wk;
        qi += it * wq; ki += it * wk;
    }
    sqU[c] = qu; skU[c] = ku; sqI[c] = qi; skI[c] = ki;
    __syncthreads();
    if (c == 0) {
        float s_ui = 0.f, s_ii = 0.f, s_iu = 0.f, s_uu = 0.f;
        for (int j = 0; j < CD; ++j) {
            s_ui += skU[j] * sqI[j];
            s_ii += skI[j] * sqI[j];
            s_iu += skI[j] * sqU[j];
            s_uu += skU[j] * sqU[j];
        }
        const float inv = 0.125f;    // 1/sqrt(KQ_DIM=64)
        s_ui *= inv; s_ii *= inv; s_iu *= inv; s_uu *= inv;
        float m0 = fmaxf(s_ui, s_ii);
        float e0 = __expf(s_ui - m0), e1 = __expf(s_ii - m0);
        float d0 = e0 + e1;
        gA[0] = e0 / d0; gA[1] = e1 / d0;       // gA_item
        float m1 = fmaxf(s_iu, s_uu);
        float f0 = __expf(s_iu - m1), f1 = __expf(s_uu - m1);
        float d1 = f0 + f1;
        gA[2] = f0 / d1; gA[3] = f1 / d1;       // gA_user
    }
}

// ---- edge pass 1: segment max (monotone-uint atomicMax) ----
__global__ __launch_bounds__(256) void k_edge_max(const int* __restrict__ src,
                                                  const int* __restrict__ dst,
                                                  const float* __restrict__ ss,
                                                  const float* __restrict__ sd,
                                                  unsigned* __restrict__ mkey, int ne) {
    int i = blockIdx.x * blockDim.x + threadIdx.x;
    int step = gridDim.x * blockDim.x;
    for (; i < ne; i += step) {
        float e = lrelu(ss[src[i]] + sd[dst[i]]);
        atomicMax(&mkey[dst[i]], f2key(e));
    }
}

// ---- edge pass 2: exp and segment sum ----
__global__ __launch_bounds__(256) void k_edge_den(const int* __restrict__ src,
                                                  const int* __restrict__ dst,
                                                  const float* __restrict__ ss,
                                                  const float* __restrict__ sd,
                                                  const unsigned* __restrict__ mkey,
                                                  float* __restrict__ den,
                                                  float* __restrict__ ex, int ne) {
    int i = blockIdx.x * blockDim.x + threadIdx.x;
    int step = gridDim.x * blockDim.x;
    for (; i < ne; i += step) {
        int d = dst[i];
        float e = lrelu(ss[src[i]] + sd[d]);
        float x = __expf(e - key2f(mkey[d]));
        ex[i] = x;
        atomicAdd(&den[d], x);
    }
}

// ---- edge pass 3: weighted scatter of 128-dim rows (wave per edge, float4 lanes) ----
__global__ __launch_bounds__(256) void k_edge_agg(const int* __restrict__ src,
                                                  const int* __restrict__ dst,
                                                  const float* __restrict__ ex,
                                                  const float* __restrict__ den,
                                                  const float* __restrict__ hsrc,
                                                  float* __restrict__ out,
                                                  const float* __restrict__ gA, int gaIdx,
                                                  int ne) {
    const int lane = threadIdx.x & 31;
    int e = (blockIdx.x * blockDim.x + threadIdx.x) >> 5;
    const int nw = (gridDim.x * blockDim.x) >> 5;
    const float g = gA[gaIdx];
    for (; e < ne; e += nw) {
        const int s = src[e], d = dst[e];
        const float a = ex[e] / fmaxf(den[d], 1e-30f) * g;
        const float4 x = *(const float4*)(hsrc + (size_t)s * IN_DIM + 4 * lane);
        float* o = out + (size_t)d * IN_DIM + 4 * lane;
        atomicAdd(o + 0, a * x.x);
        atomicAdd(o + 1, a * x.y);
        atomicAdd(o + 2, a * x.z);
        atomicAdd(o + 3, a * x.w);
    }
}

// ---- final FC, in place on d_out: f32 WMMA 16x16x4, K=128 in 32 steps ----
__global__ __launch_bounds__(256) void k_fc_wmma(float* __restrict__ io,
                                                 const float* __restrict__ W,
                                                 const float* __restrict__ bias,
                                                 int nrows) {
    __shared__ float sA[16 * IN_DIM];           // 16-row input tile (pre-FC values)
    const int rowBase = blockIdx.x * 16;
    for (int i = threadIdx.x; i < 16 * IN_DIM; i += 256) {
        int r = rowBase + (i >> 7);
        sA[i] = (r < nrows) ? io[(size_t)rowBase * IN_DIM + i] : 0.f;
    }
    __syncthreads();

    const int lane    = threadIdx.x & 31;
    const int colBase = (threadIdx.x >> 5) * 16;   // 8 waves x 16 cols = 128 outputs
    const int lrow    = lane & 15;                 // M index for A, N index for B/C
    const int khalf   = (lane >> 4) * 2;           // lanes 16-31 carry K+2,K+3

    v8f c = {0.f, 0.f, 0.f, 0.f, 0.f, 0.f, 0.f, 0.f};
    for (int k0 = 0; k0 < IN_DIM; k0 += 4) {
        v2f a, b;
        a.x = sA[lrow * IN_DIM + k0 + khalf];
        a.y = sA[lrow * IN_DIM + k0 + khalf + 1];
        // out = h @ W^T  =>  B(k,n) = W[n][k]
        b.x = W[(colBase + lrow) * IN_DIM + k0 + khalf];
        b.y = W[(colBase + lrow) * IN_DIM + k0 + khalf + 1];
        c = __builtin_amdgcn_wmma_f32_16x16x4_f32(false, a, false, b,
                                                  (short)0, c, false, false);
    }

    const float bb = bias[colBase + lrow];
    for (int r = 0; r < 8; ++r) {
        const int M = r + ((lane >> 4) << 3);     // C/D layout: lanes 16-31 hold M+8
        const int row = rowBase + M;
        if (row < nrows)
            io[(size_t)row * IN_DIM + colBase + lrow] = c[r] + bb;
    }
}

extern "C" void kernel_launch(void* const* d_in, const int* in_sizes, int n_in,
                              void* d_out, int out_size, void* d_ws, size_t ws_size,
                              hipStream_t stream) {
    const float* h_user = (const float*)d_in[0];
    const float* h_item = (const float*)d_in[1];
    const float* Wc     = (const float*)d_in[2];
    const float* Wk     = (const float*)d_in[3];
    const float* Wq     = (const float*)d_in[4];
    const float* attn_w = (const float*)d_in[5];
    const float* fcW    = (const float*)d_in[6];
    const float* fcb    = (const float*)d_in[7];
    const int* src_rates   = (const int*)d_in[8];
    const int* dst_rates   = (const int*)d_in[9];
    const int* src_similar = (const int*)d_in[10];
    const int* dst_similar = (const int*)d_in[11];
    const int* src_rated   = (const int*)d_in[12];
    const int* dst_rated   = (const int*)d_in[13];
    const int* src_follows = (const int*)d_in[14];
    const int* dst_follows = (const int*)d_in[15];

    const int NU = in_sizes[0] / IN_DIM;
    const int NI = in_sizes[1] / IN_DIM;
    const int NE = in_sizes[8];

    float* out      = (float*)d_out;
    float* out_user = out;                              // accumulates h_user_new, then FC in place
    float* out_item = out + (size_t)NU * IN_DIM;        // accumulates h_item_new

    // ---- workspace layout (floats) ----
    float* ws     = (float*)d_ws;
    float* gsumU  = ws;            // 128
    float* gsumI  = ws + 128;      // 128
    float* vs     = ws + 256;      // 128
    float* vd     = ws + 384;      // 128
    float* gA     = ws + 512;      // 4
    float* p      = ws + 1024;
    float* sus = p; p += NU;       // user scores (source-side)
    float* sud = p; p += NU;       // user scores (dest-side)
    float* sis = p; p += NI;
    float* sid = p; p += NI;
    unsigned* m_rates = (unsigned*)p; p += NI;
    float*    d_rates = p;         p += NI;
    unsigned* m_sim   = (unsigned*)p; p += NI;
    float*    d_sim   = p;         p += NI;
    unsigned* m_rated = (unsigned*)p; p += NU;
    float*    d_rated = p;         p += NU;
    unsigned* m_fol   = (unsigned*)p; p += NU;
    float*    d_fol   = p;         p += NU;
    float* exbuf = p; p += NE;     // reused per edge type (stream-sequential)
    const size_t ws_used = (size_t)(p - ws) * sizeof(float);

    // zero accumulators every call (graph-capture legal)
    hipMemsetAsync(d_out, 0, (size_t)out_size * sizeof(float), stream);
    hipMemsetAsync(d_ws, 0, ws_used, stream);

    // stage 1: folded projection vectors
    k_vsvd<<<1, 128, 0, stream>>>(Wc, attn_w, vs, vd);

    // stage 2: fused mean column-sums + per-node scores (single read of h_*)
    k_stats<<<1024, 256, 0, stream>>>(h_user, NU, sus, sud, vs, vd, gsumU);
    k_stats<<<1024, 256, 0, stream>>>(h_item, NI, sis, sid, vs, vd, gsumI);

    // stage 3: global gates
    k_ga<<<1, CD, 0, stream>>>(gsumU, gsumI, Wq, Wk, gA, 1.0f / NU, 1.0f / NI);

    // stage 4: four edge-type segment-softmax aggregations
    const int ebt = (NE + 255) / 256;          // thread-per-edge grids
    const int ebw = (NE + 7) / 8;              // wave-per-edge grids (8 waves/block)

    // rates: user -> item, gate gA[0]
    k_edge_max<<<ebt, 256, 0, stream>>>(src_rates, dst_rates, sus, sid, m_rates, NE);
    k_edge_den<<<ebt, 256, 0, stream>>>(src_rates, dst_rates, sus, sid, m_rates, d_rates, exbuf, NE);
    k_edge_agg<<<ebw, 256, 0, stream>>>(src_rates, dst_rates, exbuf, d_rates, h_user, out_item, gA, 0, NE);

    // similar: item -> item, gate gA[1]
    k_edge_max<<<ebt, 256, 0, stream>>>(src_similar, dst_similar, sis, sid, m_sim, NE);
    k_edge_den<<<ebt, 256, 0, stream>>>(src_similar, dst_similar, sis, sid, m_sim, d_sim, exbuf, NE);
    k_edge_agg<<<ebw, 256, 0, stream>>>(src_similar, dst_similar, exbuf, d_sim, h_item, out_item, gA, 1, NE);

    // rated: item -> user, gate gA[2]
    k_edge_max<<<ebt, 256, 0, stream>>>(src_rated, dst_rated, sis, sud, m_rated, NE);
    k_edge_den<<<ebt, 256, 0, stream>>>(src_rated, dst_rated, sis, sud, m_rated, d_rated, exbuf, NE);
    k_edge_agg<<<ebw, 256, 0, stream>>>(src_rated, dst_rated, exbuf, d_rated, h_item, out_user, gA, 2, NE);

    // follows: user -> user, gate gA[3]
    k_edge_max<<<ebt, 256, 0, stream>>>(src_follows, dst_follows, sus, sud, m_fol, NE);
    k_edge_den<<<ebt, 256, 0, stream>>>(src_follows, dst_follows, sus, sud, m_fol, d_fol, exbuf, NE);
    k_edge_agg<<<ebw, 256, 0, stream>>>(src_follows, dst_follows, exbuf, d_fol, h_user, out_user, gA, 3, NE);

    // stage 5: final FC in place on d_out, f32 WMMA
    const int totalRows = NU + NI;
    k_fc_wmma<<<(totalRows + 15) / 16, 256, 0, stream>>>(out, fcW, fcb, totalRows);
}